// BalSCL_8306466750559
// MI455X (gfx1250) — compile-verified
//
#include <hip/hip_runtime.h>

// ---------------- problem constants ----------------
#define DIM_   128
#define B_     4096
#define TWOB_  8192
#define C_     100
#define NCOL_  8292          // 2B + C
#define NT_    519           // column tiles of 16 -> NPAD = 8304
#define NPAD_  (NT_ * 16)
#define CSPLIT_ 4
#define INV_T  10.0f         // 1 / temperature
#define K_L2E  14.4269504088896340736f   // (1/t) * log2(e)

typedef float v2f __attribute__((ext_vector_type(2)));
typedef float v8f __attribute__((ext_vector_type(8)));
typedef unsigned int v4u __attribute__((ext_vector_type(4)));
typedef int v4i __attribute__((ext_vector_type(4)));
typedef int v8i __attribute__((ext_vector_type(8)));

// ---------------- workspace layout ----------------
__host__ __device__ constexpr size_t al256(size_t x) { return (x + 255) & ~(size_t)255; }
constexpr size_t OFF_FEATS = 0;
constexpr size_t OFF_TGT   = al256(OFF_FEATS + (size_t)NPAD_ * DIM_ * 4);
constexpr size_t OFF_WTAB  = al256(OFF_TGT   + (size_t)NPAD_ * 4);
constexpr size_t OFF_CNT   = al256(OFF_WTAB  + (size_t)NPAD_ * 8);   // float2 per col
constexpr size_t OFF_PS    = al256(OFF_CNT   + 512);
constexpr size_t OFF_PC    = al256(OFF_PS    + (size_t)CSPLIT_ * TWOB_ * 4);
constexpr size_t OFF_PD    = al256(OFF_PC    + (size_t)CSPLIT_ * TWOB_ * 4);

// ---------------- prep kernels ----------------
__global__ void prep_tgt(const int* __restrict__ targets, int* __restrict__ tgt,
                         int* __restrict__ cnt) {
  int t = blockIdx.x * 256 + threadIdx.x;
  if (t >= NPAD_) return;
  int v;
  if (t < B_)         v = targets[t];
  else if (t < TWOB_) v = targets[t - B_];
  else if (t < NCOL_) v = t - TWOB_;
  else                v = -1;                 // pad column
  tgt[t] = v;
  if (v >= 0) atomicAdd(&cnt[v], 1);          // integer atomic: deterministic
}

__global__ void prep_feats(const float* __restrict__ features,
                           const float* __restrict__ centers,
                           float* __restrict__ feats) {
  int idx = blockIdx.x * 256 + threadIdx.x;   // one float4 per thread
  if (idx >= NPAD_ * (DIM_ / 4)) return;
  int r = idx >> 5;
  int c = (idx & 31) * 4;
  float4 vv;
  if (r < B_)         vv = *(const float4*)(features + (size_t)r * 2 * DIM_ + c);
  else if (r < TWOB_) vv = *(const float4*)(features + (size_t)(r - B_) * 2 * DIM_ + DIM_ + c);
  else if (r < NCOL_) vv = *(const float4*)(centers  + (size_t)(r - TWOB_) * DIM_ + c);
  else                vv = make_float4(0.f, 0.f, 0.f, 0.f);   // zero pad rows
  *(float4*)(feats + (size_t)r * DIM_ + c) = vv;
}

// per-column reciprocal weights: {1/cnt, 1/(cnt-1)} ; pad columns -> {0,0}
__global__ void prep_wtab(const int* __restrict__ tgt, const int* __restrict__ cnt,
                          float* __restrict__ wtab) {
  int j = blockIdx.x * 256 + threadIdx.x;
  if (j >= NPAD_) return;
  int tv = tgt[j];
  float r0 = 0.f, r1 = 0.f;
  if (tv >= 0) {
    float cf = (float)cnt[tv];
    r0 = 1.0f / cf;
    r1 = 1.0f / (cf - 1.0f);   // only selected when a same-class row exists (cnt>=3)
  }
  wtab[2 * j]     = r0;
  wtab[2 * j + 1] = r1;
}

// ---------------- TDM tile DMA: 16 rows x 128 fp32, LDS row stride 132 ----------------
// D# per CDNA5 ISA 8.3/8.4: 2-D tile, data_size=4B, pad 4 DWORDs every 128 DWORDs.
__device__ __forceinline__ void tdm_load_tile(const float* gsrc, unsigned lds_byte) {
  unsigned long long ga = (unsigned long long)(uintptr_t)gsrc;
  v4u g0;
  g0.x = 1u;                                            // count=1, user descriptor
  g0.y = lds_byte;                                      // lds_addr
  g0.z = (unsigned)(ga & 0xFFFFFFFFu);                  // global_addr[31:0]
  g0.w = (unsigned)((ga >> 32) & 0x1FFFFFFu)            // global_addr[56:32]
         | (2u << 30);                                  // type = 2 (image)
  v8i g1;
  g1[0] = (int)((2u << 16)      // data_size = 4B
              | (1u << 20)      // pad_enable
              | (6u << 22)      // pad_interval: 128 DWORDs
              | (3u << 25));    // pad_amount: 4 DWORDs -> row stride 132 floats
  g1[1] = (int)(128u << 16);    // tensor_dim0 = 128 (bits 79:48)
  g1[2] = (int)(16u  << 16);    // tensor_dim1 = 16  (bits 111:80)
  g1[3] = (int)(128u << 16);    // tile_dim0 = 128   (bits 127:112)
  g1[4] = 16;                   // tile_dim1 = 16
  g1[5] = 128;                  // tensor_dim0_stride = 128 elements
  g1[6] = 0;
  g1[7] = 0;
  v4i z4 = {0, 0, 0, 0};        // groups 2/3 unused (2-D tensor)
  v8i z8 = {0, 0, 0, 0, 0, 0, 0, 0};
  __builtin_amdgcn_tensor_load_to_lds(g0, g1, z4, z4, z8, 0);  // 6-arg (clang-23) form
}

__device__ __forceinline__ unsigned lds_offset_of(const void* p) {
  // LDS aperture: flat addr[31:0] is the LDS byte offset (ISA 10.2)
  return (unsigned)(unsigned long long)(uintptr_t)p;
}

// ---------------- fused GEMM + softmax-statistics ----------------
// grid = (TWOB_/128, CSPLIT_), block = 256 (8 wave32s; each wave owns 16 rows)
__global__ __launch_bounds__(256) void bal_gemm_softmax(
    const float* __restrict__ feats, const int* __restrict__ tgt,
    const float* __restrict__ wtab,
    float* __restrict__ pS, float* __restrict__ pC, float* __restrict__ pD)
{
  __shared__ __align__(16) float ldsB[2][16 * 132];   // TDM pads rows to stride 132

  const int tid  = threadIdx.x;
  const int wv   = tid >> 5;
  const int lane = tid & 31;
  const int hlf  = lane >> 4;      // half-wave
  const int l16  = lane & 15;
  const int rbase = blockIdx.x * 128 + wv * 16;

  // A fragments: fp32 16x4 layout -> lane holds row l16, K pair (kk*4 + hlf*2)
  v2f afrag[32];
  {
    const float* arow = feats + (size_t)(rbase + l16) * DIM_ + hlf * 2;
    #pragma unroll
    for (int kk = 0; kk < 32; ++kk)
      afrag[kk] = *(const v2f*)(arow + kk * 4);
  }
  // targets of this half-wave's 8 rows (C/D layout: M = v + hlf*8)
  int ti[8];
  #pragma unroll
  for (int v = 0; v < 8; ++v) ti[v] = tgt[rbase + hlf * 8 + v];

  float accS[8], accC[8], accD[8];
  #pragma unroll
  for (int v = 0; v < 8; ++v) { accS[v] = 0.f; accC[v] = 0.f; accD[v] = 0.f; }

  const int t0 = (NT_ * (int)blockIdx.y) / CSPLIT_;
  const int t1 = (NT_ * ((int)blockIdx.y + 1)) / CSPLIT_;
  const unsigned ldsOff[2] = { lds_offset_of(&ldsB[0][0]), lds_offset_of(&ldsB[1][0]) };

  // stage first tile via the Tensor Data Mover (wave 0 issues, TENSORcnt tracks)
  if (wv == 0) {
    tdm_load_tile(feats + (size_t)t0 * 16 * DIM_, ldsOff[0]);
    __builtin_amdgcn_s_wait_tensorcnt(0);
  }
  asm volatile("" ::: "memory");
  __syncthreads();

  for (int jt = t0; jt < t1; ++jt) {
    const int buf = (jt - t0) & 1;
    if (wv == 0 && jt + 1 < t1)                      // DMA next tile while computing
      tdm_load_tile(feats + (size_t)(jt + 1) * 16 * DIM_, ldsOff[buf ^ 1]);
    if (jt + 2 < t1 && tid < 64)                     // warm GL2 for jt+2
      __builtin_prefetch(feats + (size_t)(jt + 2) * 16 * DIM_ + tid * 32, 0, 0);

    // ---- 16x16 logits tile, K = 128 via 32x V_WMMA_F32_16X16X4_F32 ----
    v8f c = {0.f, 0.f, 0.f, 0.f, 0.f, 0.f, 0.f, 0.f};
    const float* bb = &ldsB[buf][l16 * 132 + hlf * 2];   // conflict-free ds_load_b64
    #pragma unroll
    for (int kk = 0; kk < 32; ++kk) {
      v2f b = *(const v2f*)(bb + kk * 4);
      c = __builtin_amdgcn_wmma_f32_16x16x4_f32(
              false, afrag[kk], false, b, (short)0, c, false, false);
    }

    // ---- fused epilogue: one column j per lane, 8 rows per lane ----
    const int j  = jt * 16 + l16;
    const v2f rw = *(const v2f*)(wtab + 2 * j);     // {1/cnt, 1/(cnt-1)}
    const int tj = tgt[j];
    const bool jv = (tj >= 0);
    #pragma unroll
    for (int v = 0; v < 8; ++v) {
      const int   i    = rbase + hlf * 8 + v;
      const bool  ok   = jv && (j != i);            // logits_mask (diag + pad)
      const bool  same = ok && (tj == ti[v]);       // positive mask bit
      const float e    = ok ? __builtin_amdgcn_exp2f(c[v] * K_L2E) : 0.0f; // 1 mul + v_exp_f32
      accS[v] += e * (same ? rw.y : rw.x);          // no division
      accC[v] += same ? 1.0f : 0.0f;
      accD[v] += same ? c[v] : 0.0f;                // raw dot; x (1/t) deferred
    }
    if (wv == 0) __builtin_amdgcn_s_wait_tensorcnt(0);
    asm volatile("" ::: "memory");
    __syncthreads();
  }

  // reduce across the 16 lanes of each half-wave, write deterministic partials
  #pragma unroll
  for (int v = 0; v < 8; ++v) {
    float s = accS[v], cc = accC[v], d = accD[v];
    #pragma unroll
    for (int m = 1; m < 16; m <<= 1) {
      s  += __shfl_xor(s,  m, 32);
      cc += __shfl_xor(cc, m, 32);
      d  += __shfl_xor(d,  m, 32);
    }
    if (l16 == 0) {
      const int i = rbase + hlf * 8 + v;
      const size_t o = (size_t)blockIdx.y * TWOB_ + i;
      pS[o] = s; pC[o] = cc; pD[o] = d;
    }
  }
}

// ---------------- deterministic final reduction ----------------
__global__ __launch_bounds__(1024) void bal_finalize(
    const float* __restrict__ pS, const float* __restrict__ pC,
    const float* __restrict__ pD, float* __restrict__ out)
{
  __shared__ float red[1024];
  float acc = 0.f;
  for (int i = threadIdx.x; i < TWOB_; i += 1024) {
    float s = 0.f, c = 0.f, d = 0.f;
    #pragma unroll
    for (int sp = 0; sp < CSPLIT_; ++sp) {
      const size_t o = (size_t)sp * TWOB_ + i;
      s += pS[o]; c += pC[o]; d += pD[o];
    }
    acc += logf(s) - (INV_T * d) / c;   // loss_i = log(sumexp) - maskdot/masksum
  }
  red[threadIdx.x] = acc;
  __syncthreads();
  for (int st = 512; st > 0; st >>= 1) {
    if ((int)threadIdx.x < st) red[threadIdx.x] += red[threadIdx.x + st];
    __syncthreads();
  }
  if (threadIdx.x == 0) out[0] = red[0] / (float)TWOB_;
}

// ---------------- host launch ----------------
extern "C" void kernel_launch(void* const* d_in, const int* in_sizes, int n_in,
                              void* d_out, int out_size, void* d_ws, size_t ws_size,
                              hipStream_t stream) {
  const float* centers1 = (const float*)d_in[0];   // [100,128] f32
  const float* features = (const float*)d_in[1];   // [4096,2,128] f32
  const int*   targets  = (const int*)d_in[2];     // [4096] i32
  (void)in_sizes; (void)n_in; (void)out_size; (void)ws_size;

  char* ws = (char*)d_ws;
  float* feats = (float*)(ws + OFF_FEATS);
  int*   tgt   = (int*)  (ws + OFF_TGT);
  float* wtab  = (float*)(ws + OFF_WTAB);
  int*   cnt   = (int*)  (ws + OFF_CNT);
  float* pS    = (float*)(ws + OFF_PS);
  float* pC    = (float*)(ws + OFF_PC);
  float* pD    = (float*)(ws + OFF_PD);

  (void)hipMemsetAsync(cnt, 0, 512, stream);

  const int g1 = (NPAD_ + 255) / 256;                       // 33
  prep_tgt<<<g1, 256, 0, stream>>>(targets, tgt, cnt);
  const int g2 = (NPAD_ * (DIM_ / 4) + 255) / 256;          // 1038
  prep_feats<<<g2, 256, 0, stream>>>(features, centers1, feats);
  prep_wtab<<<g1, 256, 0, stream>>>(tgt, cnt, wtab);

  bal_gemm_softmax<<<dim3(TWOB_ / 128, CSPLIT_), 256, 0, stream>>>(
      feats, tgt, wtab, pS, pC, pD);

  bal_finalize<<<1, 1024, 0, stream>>>(pS, pC, pD, (float*)d_out);
}